// SimpleRelativeAttention_49151605736045
// MI455X (gfx1250) — compile-verified
//
#include <hip/hip_runtime.h>
#include <hip/hip_bf16.h>

// ---------------------------------------------------------------------------
// SimpleRelativeAttention, MI455X (gfx1250, wave32, WMMA).
//
// The reference's final einsum 'bnqk,bnqe->bnqe' contracts attn's k-axis
// alone, so out = v * rowsum(softmax(...)) = v (rows of softmax sum to 1).
// Hence:  out = x @ (w_qkv[:,2C:3C] @ w_proj) + (b_qkv[2C:3C] @ w_proj + b_proj)
// rel_table / rel_index cannot affect the output.
// Implemented as two f32 WMMA GEMMs (V_WMMA_F32_16X16X4_F32) + a tiny bias
// fusion kernel. f32 matrix path keeps reference precision exactly.
// ---------------------------------------------------------------------------

typedef float v2f __attribute__((ext_vector_type(2)));
typedef float v8f __attribute__((ext_vector_type(8)));

#define TILE_M 128
#define TILE_N 128
#define TILE_K 32
#define LDS_A_STRIDE 36   // 128x32 A tile, padded: conflict-free float2 reads, 16B rows
#define LDS_B_STRIDE 136  // 32x128 B tile, padded: half-waves hit disjoint bank ranges

// C[M,N] = A[M,K](lda) @ B[K,N](ldb) + bias[N]   (all dims multiples of tiles)
__global__ __launch_bounds__(256)
void gemm_f32_wmma(const float* __restrict__ A, int lda,
                   const float* __restrict__ B, int ldb,
                   float* __restrict__ C, int ldc,
                   const float* __restrict__ bias,
                   int K)
{
    __shared__ float sA[TILE_M * LDS_A_STRIDE];
    __shared__ float sB[TILE_K * LDS_B_STRIDE];

    const int t    = threadIdx.x;
    const int lane = t & 31;
    const int wave = t >> 5;       // 8 waves
    const int wm   = wave & 3;     // 4 waves along M: 32 rows each
    const int wn   = wave >> 2;    // 2 waves along N: 64 cols each
    const int half = lane >> 4;    // 0 | 1 (which 16-lane half)
    const int l16  = lane & 15;

    const int m0 = blockIdx.y * TILE_M;
    const int n0 = blockIdx.x * TILE_N;

    v8f acc[2][4] = {};            // 2x4 grid of 16x16 f32 tiles per wave

    // Global->LDS staging coordinates (256 threads, float4 each).
    const int a_r  = t >> 3;        // 0..31 : A rows per pass (4 passes -> 128)
    const int a_c  = (t & 7) * 4;   // 0..28 : A k-columns
    const int bB_c = (t & 31) * 4;  // 0..124: B n-columns
    const int bB_r = t >> 5;        // 0..7  : B k-rows per pass (4 passes -> 32)

    for (int k0 = 0; k0 < K; k0 += TILE_K) {
        // Stage A tile: 128 rows x 32 k
        #pragma unroll
        for (int p = 0; p < 4; ++p) {
            const int r = p * 32 + a_r;
            const float4 v = *(const float4*)(A + (size_t)(m0 + r) * lda + k0 + a_c);
            *(float4*)(sA + r * LDS_A_STRIDE + a_c) = v;
        }
        // Stage B tile: 32 k-rows x 128 n
        #pragma unroll
        for (int p = 0; p < 4; ++p) {
            const int r = p * 8 + bB_r;
            const float4 v = *(const float4*)(B + (size_t)(k0 + r) * ldb + n0 + bB_c);
            *(float4*)(sB + r * LDS_B_STRIDE + bB_c) = v;
        }
        __syncthreads();

        #pragma unroll
        for (int kk = 0; kk < TILE_K; kk += 4) {
            // A fragment (16x4 f32): lane holds row M=l16, K = kk + 2*half + {0,1}
            v2f afrag[2];
            #pragma unroll
            for (int mt = 0; mt < 2; ++mt) {
                const float* p = sA + (wm * 32 + mt * 16 + l16) * LDS_A_STRIDE
                                    + kk + 2 * half;
                afrag[mt] = *(const v2f*)p;   // 8B aligned (even k, 144B rows)
            }
            // B fragment (4x16 f32): lane holds col N=l16, K = kk + 2*half + {0,1}
            v2f bfrag[4];
            #pragma unroll
            for (int nt = 0; nt < 4; ++nt) {
                const float* p = sB + (kk + 2 * half) * LDS_B_STRIDE
                                    + wn * 64 + nt * 16 + l16;
                v2f b;
                b.x = p[0];
                b.y = p[LDS_B_STRIDE];
                bfrag[nt] = b;
            }
            #pragma unroll
            for (int mt = 0; mt < 2; ++mt)
                #pragma unroll
                for (int nt = 0; nt < 4; ++nt)
                    acc[mt][nt] = __builtin_amdgcn_wmma_f32_16x16x4_f32(
                        /*neg_a=*/false, afrag[mt],
                        /*neg_b=*/false, bfrag[nt],
                        /*c_mod=*/(short)0, acc[mt][nt],
                        /*reuse_a=*/false, /*reuse_b=*/false);
        }
        __syncthreads();
    }

    // Epilogue. C/D layout: VGPR v -> M = v + 8*half, N = l16.
    #pragma unroll
    for (int mt = 0; mt < 2; ++mt) {
        #pragma unroll
        for (int nt = 0; nt < 4; ++nt) {
            const int col = n0 + wn * 64 + nt * 16 + l16;
            const float bv = bias ? bias[col] : 0.0f;
            #pragma unroll
            for (int v = 0; v < 8; ++v) {
                const int row = m0 + wm * 32 + mt * 16 + v + 8 * half;
                C[(size_t)row * ldc + col] = acc[mt][nt][v] + bv;
            }
        }
    }
}

// b_f[j] = sum_k b_qkv_v[k] * w_proj[k, j] + b_proj[j]
__global__ __launch_bounds__(256)
void fuse_bias(const float* __restrict__ b_qkv_v,
               const float* __restrict__ w_proj,
               const float* __restrict__ b_proj,
               float* __restrict__ b_f, int K, int N)
{
    const int j = blockIdx.x * blockDim.x + threadIdx.x;
    if (j >= N) return;
    float s = b_proj[j];
    for (int k = 0; k < K; ++k)
        s += b_qkv_v[k] * w_proj[(size_t)k * N + j];
    b_f[j] = s;
}

extern "C" void kernel_launch(void* const* d_in, const int* in_sizes, int n_in,
                              void* d_out, int out_size, void* d_ws, size_t ws_size,
                              hipStream_t stream)
{
    (void)in_sizes; (void)n_in; (void)out_size; (void)ws_size;

    const float* x      = (const float*)d_in[0];   // [4,1024,1024]
    const float* w_qkv  = (const float*)d_in[1];   // [1024,3072]
    const float* b_qkv  = (const float*)d_in[2];   // [3072]
    const float* w_proj = (const float*)d_in[3];   // [1024,1024]
    const float* b_proj = (const float*)d_in[4];   // [1024]
    // d_in[5] rel_table, d_in[6] rel_index: provably dead (softmax rows sum to 1
    // and the reference einsum contracts attn's k-axis alone).

    float* out = (float*)d_out;                    // [4096,1024] f32

    const int C = 1024;
    const int M = 4096;                            // B*N = 4*1024

    float* W_f = (float*)d_ws;                     // [1024,1024] fused weight (4 MB)
    float* b_f = W_f + (size_t)C * C;              // [1024] fused bias

    dim3 blk(256);

    // 1) W_f = w_qkv[:, 2C:3C] @ w_proj        (1024^3 GEMM)
    dim3 g1(C / TILE_N, C / TILE_M);
    gemm_f32_wmma<<<g1, blk, 0, stream>>>(w_qkv + 2 * C, 3 * C,
                                          w_proj, C,
                                          W_f, C, nullptr, C);

    // 2) b_f = b_qkv[2C:3C] @ w_proj + b_proj
    fuse_bias<<<dim3((C + 255) / 256), blk, 0, stream>>>(b_qkv + 2 * C, w_proj,
                                                         b_proj, b_f, C, C);

    // 3) out = x @ W_f + b_f                   (4096x1024x1024 GEMM)
    dim3 g2(C / TILE_N, M / TILE_M);
    gemm_f32_wmma<<<g2, blk, 0, stream>>>(x, C,
                                          W_f, C,
                                          out, C, b_f, C);
}